// DigitCapsules_77781857730739
// MI455X (gfx1250) — compile-verified
//
#include <hip/hip_runtime.h>

// Problem constants (from reference)
#define B_  512
#define R_  1152
#define C_  10
#define O_  16
#define I_  8
#define EPS_ 1e-8f

// Tiling
#define BT      16               // batch tile = WMMA N
#define RCHUNKS 16
#define RPER    (R_ / RCHUNKS)   // 72 r per block
#define WAVES   8
#define BLOCK   (WAVES * 32)
#define CO_     (C_ * O_)        // 160
#define XSL     (RPER * I_)      // floats per b' slice in LDS x tile (576)
#define XCH     (BT * XSL)       // x tile floats (9216 = 36 KB)

typedef __attribute__((ext_vector_type(2))) float v2f;
typedef __attribute__((ext_vector_type(4))) float v4f;
typedef __attribute__((ext_vector_type(8))) float v8f;

// ---------------------------------------------------------------------------
// Kernel A: one routing pass. For each (batch-tile, r-chunk):
//   - async-stage x tile (16 b' x 72 r x 8 i) into LDS (ASYNCcnt path),
//   - u_hat[c,o,b'] via V_WMMA_F32_16X16X4_F32 (per r: 10 tiles x 2 K-steps),
//   - logits[b',c] = u_hat . vsum  (vsum = sum of previous v_j; 0 on iter 0),
//   - c_ij = softmax_c(logits), s[b,c,o] += c_ij * u_hat (LDS + global atomics)
// ---------------------------------------------------------------------------
__global__ __launch_bounds__(BLOCK, 1)
void caps_route_partial(const float* __restrict__ x,      // [B,R,I]
                        const float* __restrict__ Wt,     // [R,C,O,I]
                        const float* __restrict__ vsum,   // [B,C,O]
                        float* __restrict__ s_glob) {     // [B,C,O]
    __shared__ __align__(16) float x_sh[XCH];   // 36 KB staged x tile
    __shared__ float s_sh[BT * CO_];            // 10 KB block s accumulator

    const int tid   = threadIdx.x;
    const int lane  = tid & 31;
    const int wave  = tid >> 5;
    const int lmod  = lane & 15;       // b' column / A row (o)
    const int lhalf = lane >> 4;       // 0: o=0..7 / K lo half, 1: o=8..15 / hi

    const int b0    = blockIdx.y * BT;
    const int bp    = b0 + lmod;                 // this lane's batch element
    const int rbase = blockIdx.x * RPER;

    // ---- async copy x tile -> LDS (16B per lane per op, fully coalesced) ----
    // 2304 float4 units = 9 per thread, uniform (no divergence around asyncs).
    {
        const int UNITS = XCH / 4;               // 2304
        #pragma unroll
        for (int u = tid; u < UNITS; u += BLOCK) {
            const int bb  = u / (XSL / 4);       // which b' slice
            const int off = (u % (XSL / 4)) * 4; // float offset within slice
            const float* gp = x + ((size_t)(b0 + bb) * R_ + rbase) * I_ + off;
            const unsigned lo = (unsigned)(size_t)&x_sh[bb * XSL + off];
            asm volatile("global_load_async_to_lds_b128 %0, %1, off"
                         :: "v"(lo), "v"(gp) : "memory");
        }
    }

    // zero LDS accumulator while the async copy is in flight
    for (int i = tid; i < BT * CO_; i += BLOCK) s_sh[i] = 0.0f;

    // Private vsum in D-matrix layout: vsp[c][v] = vsum[bp][c][lhalf*8+v]
    float vsp[C_][8];
    {
        const float* vb = vsum + (size_t)bp * CO_ + lhalf * 8;
        #pragma unroll
        for (int c = 0; c < C_; ++c) {
            v4f t0 = *(const v4f*)(vb + c * O_);
            v4f t1 = *(const v4f*)(vb + c * O_ + 4);
            #pragma unroll
            for (int j = 0; j < 4; ++j) { vsp[c][j] = t0[j]; vsp[c][4 + j] = t1[j]; }
        }
    }

    // Private s accumulator (same layout as D)
    float sp[C_][8];
    #pragma unroll
    for (int c = 0; c < C_; ++c)
        #pragma unroll
        for (int v = 0; v < 8; ++v) sp[c][v] = 0.0f;

    asm volatile("s_wait_asynccnt 0" ::: "memory");
    __syncthreads();                              // x tile + zeroed s_sh visible

    for (int k = 0; k < RPER / WAVES; ++k) {
        const int rl = wave + k * WAVES;          // r local to chunk
        const int r  = rbase + rl;

        // B operand from LDS: X_r is 8x16 -> two 4x16 f32 B-matrices.
        // v0 holds K = kk*4 + lhalf, v1 holds K = kk*4 + 2 + lhalf; N = lmod.
        const float* xr = x_sh + lmod * XSL + rl * I_;
        v2f bm0, bm1;
        bm0.x = xr[lhalf];          bm0.y = xr[2 + lhalf];
        bm1.x = xr[4 + lhalf];      bm1.y = xr[6 + lhalf];

        const float* wr = Wt + (size_t)r * (CO_ * I_);
        __builtin_prefetch(wr + (size_t)WAVES * (CO_ * I_), 0, 3);  // next r

        // u_hat tiles: D[c] = W[r,c] (16x8) x X_r (8x16), K=8 as 2 WMMA steps
        v8f d[C_];
        #pragma unroll
        for (int c = 0; c < C_; ++c) {
            // A 16x4 layout: lane row M = lmod(=o), v0/v1 = K 2*lhalf, 2*lhalf+1
            const float* wrc = wr + c * (O_ * I_) + lmod * I_ + 2 * lhalf;
            v2f a0 = *(const v2f*)(wrc);        // i = 2*lhalf + {0,1}
            v2f a1 = *(const v2f*)(wrc + 4);    // i = 4 + 2*lhalf + {0,1}
            v8f acc = {};
            acc = __builtin_amdgcn_wmma_f32_16x16x4_f32(
                      false, a0, false, bm0, (short)0, acc, false, false);
            acc = __builtin_amdgcn_wmma_f32_16x16x4_f32(
                      false, a1, false, bm1, (short)0, acc, false, false);
            d[c] = acc;
        }

        // logits[c] = u_hat[bp,r,c,:] . vsum[bp,c,:]  (join o-halves via xor-16)
        float logit[C_];
        #pragma unroll
        for (int c = 0; c < C_; ++c) {
            float p = 0.0f;
            #pragma unroll
            for (int v = 0; v < 8; ++v) p += d[c][v] * vsp[c][v];
            logit[c] = p + __shfl_xor(p, 16, 32);
        }

        // softmax over capsules
        float m = logit[0];
        #pragma unroll
        for (int c = 1; c < C_; ++c) m = fmaxf(m, logit[c]);
        float e[C_]; float sum = 0.0f;
        #pragma unroll
        for (int c = 0; c < C_; ++c) { e[c] = __expf(logit[c] - m); sum += e[c]; }
        const float inv = 1.0f / sum;

        // s += c_ij * u_hat (fully in registers)
        #pragma unroll
        for (int c = 0; c < C_; ++c) {
            const float cij = e[c] * inv;
            #pragma unroll
            for (int v = 0; v < 8; ++v) sp[c][v] += cij * d[c][v];
        }
    }

    // Reduce private -> LDS (ds_add_f32), then LDS -> global atomics
    {
        const int base = lmod * CO_ + lhalf * 8;
        #pragma unroll
        for (int c = 0; c < C_; ++c)
            #pragma unroll
            for (int v = 0; v < 8; ++v)
                atomicAdd(&s_sh[base + c * O_ + v], sp[c][v]);
    }
    __syncthreads();
    float* sg = s_glob + (size_t)b0 * CO_;
    for (int i = tid; i < BT * CO_; i += BLOCK) atomicAdd(&sg[i], s_sh[i]);
}

// ---------------------------------------------------------------------------
// Kernel B: squash s -> v, vsum += v, re-zero s; final iter writes d_out.
// One block per batch element, 160 threads = (c,o).
// ---------------------------------------------------------------------------
__global__ __launch_bounds__(CO_)
void caps_squash(float* __restrict__ s_glob,   // [B,C,O]
                 float* __restrict__ vsum,     // [B,C,O]
                 float* __restrict__ out,      // [B,C,O]
                 int write_out) {
    __shared__ float sq_sh[CO_];
    __shared__ float red[C_];
    const int b   = blockIdx.x;
    const int tid = threadIdx.x;            // c*16 + o
    const size_t idx = (size_t)b * CO_ + tid;

    const float val = s_glob[idx];
    sq_sh[tid] = val * val;
    __syncthreads();
    if (tid < C_) {
        float sq = 0.0f;
        #pragma unroll
        for (int o = 0; o < O_; ++o) sq += sq_sh[tid * O_ + o];
        red[tid] = sq;
    }
    __syncthreads();
    const float sq = red[tid >> 4];
    const float vj = val * sq / ((1.0f + sq) * sqrtf(sq + EPS_));

    vsum[idx] += vj;
    s_glob[idx] = 0.0f;                     // ready for next routing pass
    if (write_out) out[idx] = vj;
}

__global__ void zero_ws_kernel(float* p, int n) {
    int i = blockIdx.x * blockDim.x + threadIdx.x;
    if (i < n) p[i] = 0.0f;
}

// ---------------------------------------------------------------------------
extern "C" void kernel_launch(void* const* d_in, const int* in_sizes, int n_in,
                              void* d_out, int out_size, void* d_ws, size_t ws_size,
                              hipStream_t stream) {
    const float* x  = (const float*)d_in[0];   // [512,1152,8]
    const float* Wt = (const float*)d_in[1];   // [1,1152,10,16,8]
    float* out = (float*)d_out;                // [512,10,16]

    float* s_glob = (float*)d_ws;              // [B,C,O]
    float* vsum   = s_glob + (size_t)B_ * CO_; // [B,C,O]

    const int nz = 2 * B_ * CO_;
    zero_ws_kernel<<<(nz + 255) / 256, 256, 0, stream>>>(s_glob, nz);

    dim3 gridA(RCHUNKS, B_ / BT);              // 16 x 32 = 512 workgroups
    for (int it = 0; it < 3; ++it) {
        caps_route_partial<<<gridA, BLOCK, 0, stream>>>(x, Wt, vsum, s_glob);
        caps_squash<<<B_, CO_, 0, stream>>>(s_glob, vsum, out, it == 2 ? 1 : 0);
    }
}